// SupPixPool_34505767256231
// MI455X (gfx1250) — compile-verified
//
#include <hip/hip_runtime.h>
#include <cmath>

// ---------------------------------------------------------------------------
// Fused batched superpixel segment-max pooling for MI455X (gfx1250).
// img:[4,64,512,512] f32, spx:[4,512,512] i32 in [0,1024) -> out:[4,64,1024] f32
// Memory-bound (261 MB reads, ~11us floor @ 23.3 TB/s). Strategy:
//   * LDS-privatized [32ch x 1024seg] f32 max table (128 KB of the 320 KB WGP LDS)
//   * ds_max_num_f32 no-return LDS float atomics (bank = label mod 64 -> spread)
//   * labels staged by GLOBAL_LOAD_ASYNC_TO_LDS_B128 (double-buffered, ASYNCcnt)
//   * cross-WG merge via global_atomic_max_num_f32 into L2-resident 1 MB output
// Round-2 tuning: no "memory" clobber on the pure-LDS / pure-out atomics so the
// scheduler can keep ~16 b128 loads in flight per wave; waits stay fenced.
// ---------------------------------------------------------------------------

namespace {
constexpr int Bv   = 4;
constexpr int Cv   = 64;
constexpr int HWv  = 512 * 512;        // 262144 pixels per image
constexpr int Kv   = 1024;             // segments
constexpr int CSUB = 32;               // channels per workgroup
constexpr int CBLK = Cv / CSUB;        // 2 channel blocks
constexpr int GP   = 32;               // pixel blocks per image
constexpr int PPB  = HWv / GP;         // 8192 pixels per workgroup
constexpr int BS   = 256;              // threads per workgroup (8 wave32)
constexpr int VEC  = 4;                // pixels per thread per chunk (b128)
constexpr int CHUNK = BS * VEC;        // 1024 pixels per chunk
constexpr int NCHUNK = PPB / CHUNK;    // 8 chunks
}

__global__ __launch_bounds__(BS)
void spx_init_kernel(float* __restrict__ out) {
  // out has B*C*K = 262144 floats = 65536 float4; grid covers it exactly.
  const int i = blockIdx.x * BS + threadIdx.x;
  reinterpret_cast<float4*>(out)[i] =
      make_float4(-INFINITY, -INFINITY, -INFINITY, -INFINITY);
}

__global__ __launch_bounds__(BS)
void spx_pool_kernel(const float* __restrict__ img,
                     const int*   __restrict__ spx,
                     float*       __restrict__ out) {
  // 128 KB max table, [channel][segment] so DS bank = segment mod 64.
  __shared__ __align__(16) float table[CSUB * Kv];
  // Double-buffered label staging, filled by async-to-LDS copies.
  __shared__ __align__(16) int   labs[2][CHUNK];

  const int tid = threadIdx.x;
  const unsigned gid = blockIdx.x;
  const int g  = gid % GP;                   // pixel block
  const int cb = (gid / GP) % CBLK;          // channel block
  const int b  = gid / (GP * CBLK);          // batch
  const int p0 = g * PPB;
  const int cbase = cb * CSUB;

  // ---- init LDS table to -inf (b128 stores) --------------------------------
  {
    float4* t4 = reinterpret_cast<float4*>(table);
    const float4 ninf = make_float4(-INFINITY, -INFINITY, -INFINITY, -INFINITY);
    for (int i = tid; i < (CSUB * Kv) / 4; i += BS) t4[i] = ninf;
  }

  const int* glab = spx + (size_t)b * HWv + p0;
  const unsigned tbase = (unsigned)(unsigned long long)(&table[0]);

  // ---- prologue: async-stage chunk 0 labels into LDS buffer 0 --------------
  {
    const unsigned ldst = (unsigned)(unsigned long long)(&labs[0][tid * VEC]);
    const int* gsrc = glab + tid * VEC;
    asm volatile("global_load_async_to_lds_b128 %0, %1, off"
                 :: "v"(ldst), "v"(gsrc) : "memory");
  }

  __syncthreads();   // table init visible to all waves before any ds_max

  // ---- main streaming loop -------------------------------------------------
  for (int k = 0; k < NCHUNK; ++k) {
    const int buf = k & 1;

    // Prefetch labels for chunk k+1 into the other buffer (distance-1 async).
    if (k + 1 < NCHUNK) {
      const unsigned ldst =
          (unsigned)(unsigned long long)(&labs[buf ^ 1][tid * VEC]);
      const int* gsrc = glab + (k + 1) * CHUNK + tid * VEC;
      asm volatile("global_load_async_to_lds_b128 %0, %1, off"
                   :: "v"(ldst), "v"(gsrc) : "memory");
      asm volatile("s_wait_asynccnt 0x1" ::: "memory");  // chunk k copy done
    } else {
      asm volatile("s_wait_asynccnt 0x0" ::: "memory");
    }

    // Each thread reads only its own 16B label slot -> no barrier needed.
    const int4 L = *reinterpret_cast<const int4*>(&labs[buf][tid * VEC]);

    unsigned a0 = tbase + 4u * (unsigned)L.x;
    unsigned a1 = tbase + 4u * (unsigned)L.y;
    unsigned a2 = tbase + 4u * (unsigned)L.z;
    unsigned a3 = tbase + 4u * (unsigned)L.w;

    const int base = k * CHUNK + tid * VEC;  // pixel within WG range
    const float* ip =
        img + (size_t)(b * Cv + cbase) * HWv + (size_t)(p0 + base);

    // No "memory" clobber on ds_max: it only touches the LDS table, which the
    // compiler never reads until after the fenced s_wait_dscnt + barrier.
    // This lets the scheduler hoist many b128 loads ahead of the DS atomics.
    #pragma unroll 16
    for (int c = 0; c < CSUB; ++c) {
      // Coalesced: lanes cover 512B contiguous per channel row.
      const float4 v = *reinterpret_cast<const float4*>(ip + (size_t)c * HWv);
      asm volatile("ds_max_num_f32 %0, %1" :: "v"(a0), "v"(v.x));
      asm volatile("ds_max_num_f32 %0, %1" :: "v"(a1), "v"(v.y));
      asm volatile("ds_max_num_f32 %0, %1" :: "v"(a2), "v"(v.z));
      asm volatile("ds_max_num_f32 %0, %1" :: "v"(a3), "v"(v.w));
      a0 += 4u * Kv; a1 += 4u * Kv; a2 += 4u * Kv; a3 += 4u * Kv;
    }
  }

  // Inline-asm DS atomics are invisible to the compiler's waitcnt pass:
  // drain DScnt explicitly before the workgroup barrier.
  asm volatile("s_wait_dscnt 0x0" ::: "memory");
  __syncthreads();

  // ---- flush: merge LDS table into global output with f32 max atomics ------
  // table entry e = c*K + seg maps to out[(b*C + cbase)*K + e] contiguously.
  // Atomics only touch `out` (never read in-kernel): no clobber, and use the
  // instruction offset field instead of VALU address adds.
  float* ob = out + (size_t)(b * Cv + cbase) * Kv;
  for (int i = tid; i < (CSUB * Kv) / 4; i += BS) {
    const float4 v = *reinterpret_cast<const float4*>(&table[i * 4]);
    float* op = ob + i * 4;
    asm volatile("global_atomic_max_num_f32 %0, %1, off"
                 :: "v"(op), "v"(v.x));
    asm volatile("global_atomic_max_num_f32 %0, %1, off offset:4"
                 :: "v"(op), "v"(v.y));
    asm volatile("global_atomic_max_num_f32 %0, %1, off offset:8"
                 :: "v"(op), "v"(v.z));
    asm volatile("global_atomic_max_num_f32 %0, %1, off offset:12"
                 :: "v"(op), "v"(v.w));
  }
  // STOREcnt for the no-return atomics drains at s_endpgm (implicit wait-idle).
}

extern "C" void kernel_launch(void* const* d_in, const int* in_sizes, int n_in,
                              void* d_out, int out_size, void* d_ws, size_t ws_size,
                              hipStream_t stream) {
  (void)in_sizes; (void)n_in; (void)d_ws; (void)ws_size; (void)out_size;
  const float* img = (const float*)d_in[0];
  const int*   spx = (const int*)d_in[1];
  float*       out = (float*)d_out;

  // 1) out = -inf (segment_max identity; empty segments stay -inf)
  spx_init_kernel<<<(Bv * Cv * Kv) / (BS * 4), BS, 0, stream>>>(out);
  // 2) 256 WGs: B(4) x channel-blocks(2) x pixel-blocks(32)
  spx_pool_kernel<<<Bv * CBLK * GP, BS, 0, stream>>>(img, spx, out);
}